// MambaBinaryClassifier_3599182594853
// MI455X (gfx1250) — compile-verified
//
#include <hip/hip_runtime.h>
#include <hip/hip_bf16.h>
#include <math.h>

// Model constants (fixed by the reference)
#define DM 2      // D_MODEL
#define DS 16     // D_STATE
#define DC 4      // D_CONV
#define DI 4      // D_INNER
#define NPROJ 512
#define NCHUNK 128
#define KSPLIT 16 // K-slices for the WMMA GEMM (occupancy)

typedef __attribute__((ext_vector_type(2))) float v2f;
typedef __attribute__((ext_vector_type(8))) float v8f;

__device__ __forceinline__ float silu_f(float v) {
    return v * (1.0f / (1.0f + expf(-v)));
}
__device__ __forceinline__ float softplus_f(float v) {
    return v > 20.0f ? v : log1pf(expf(v));
}

// ---------------------------------------------------------------------------
// Per-lane constant context for the scan kernels.
// Wave32 per (batch, chunk). Lane layout: s = lane&15, dgroup = lane>>4.
// Each lane owns state elements (d0 = dgroup, s) and (d1 = dgroup+2, s).
// ---------------------------------------------------------------------------
struct LaneCtx {
    float ipw[DI][2];   // in_proj rows 0..3  (x path)
    float zw[DI][2];    // in_proj rows 4..7  (z path)
    float cw[DC][DI];   // conv_w[k][d]
    float cb[DI];       // conv_b
    float xp_dt[DI];    // x_proj_w row 0 (dt_rank = 1)
    float xp_B[DI];     // x_proj_w row 1+s
    float xp_C[DI];     // x_proj_w row 17+s
    float dtw0, dtb0, dtw1, dtb1;
    float A0, A1;       // A[d][s] = -exp(A_log[d][s])
    int d0, d1, s;
};

__device__ __forceinline__ void lane_init(LaneCtx& c, int lane,
        const float* __restrict__ in_proj_w, const float* __restrict__ conv_w,
        const float* __restrict__ conv_b,    const float* __restrict__ x_proj_w,
        const float* __restrict__ dt_proj_w, const float* __restrict__ dt_proj_b,
        const float* __restrict__ A_log)
{
    c.s = lane & 15;
    const int dg = lane >> 4;
    c.d0 = dg; c.d1 = dg + 2;
#pragma unroll
    for (int d = 0; d < DI; ++d) {
        c.ipw[d][0] = in_proj_w[d*DM + 0];
        c.ipw[d][1] = in_proj_w[d*DM + 1];
        c.zw[d][0]  = in_proj_w[(DI + d)*DM + 0];
        c.zw[d][1]  = in_proj_w[(DI + d)*DM + 1];
        c.cb[d]     = conv_b[d];
#pragma unroll
        for (int k = 0; k < DC; ++k) c.cw[k][d] = conv_w[k*DI + d];
    }
#pragma unroll
    for (int j = 0; j < DI; ++j) {
        c.xp_dt[j] = x_proj_w[j];
        c.xp_B[j]  = x_proj_w[(1 + c.s)*DI + j];
        c.xp_C[j]  = x_proj_w[(1 + DS + c.s)*DI + j];
    }
    c.dtw0 = dt_proj_w[c.d0]; c.dtb0 = dt_proj_b[c.d0];
    c.dtw1 = dt_proj_w[c.d1]; c.dtb1 = dt_proj_b[c.d1];
    c.A0 = -expf(A_log[c.d0*DS + c.s]);
    c.A1 = -expf(A_log[c.d1*DS + c.s]);
}

// Conv history at a chunk boundary depends only on x (not on the scan state),
// so each chunk can rebuild it directly: hist[d][j] = xs(t0-3+j), zero-padded.
__device__ __forceinline__ void init_hist(const LaneCtx& c,
        const float* __restrict__ xb0, const float* __restrict__ xb1,
        int t0, float hist[DI][3])
{
#pragma unroll
    for (int j = 0; j < 3; ++j) {
        const int t = t0 - 3 + j;
        const bool valid = (t >= 0);
        const float x0 = valid ? xb0[t] : 0.0f;
        const float x1 = valid ? xb1[t] : 0.0f;
#pragma unroll
        for (int d = 0; d < DI; ++d)
            hist[d][j] = valid ? (c.ipw[d][0]*x0 + c.ipw[d][1]*x1) : 0.0f;
    }
}

// Per-timestep frontend: in_proj -> causal conv -> SiLU -> x_proj -> dt.
template <bool NEED_Z>
__device__ __forceinline__ void step_front(const LaneCtx& c, float x0, float x1,
        float hist[DI][3], float xcv[DI], float zsil[DI],
        float& dtv0, float& dtv1, float& Bs, float& Cs)
{
#pragma unroll
    for (int d = 0; d < DI; ++d) {
        const float xs = c.ipw[d][0]*x0 + c.ipw[d][1]*x1;
        float xcr = c.cb[d] + c.cw[0][d]*hist[d][0] + c.cw[1][d]*hist[d][1]
                            + c.cw[2][d]*hist[d][2] + c.cw[3][d]*xs;
        hist[d][0] = hist[d][1]; hist[d][1] = hist[d][2]; hist[d][2] = xs;
        xcv[d] = silu_f(xcr);
        if (NEED_Z) zsil[d] = silu_f(c.zw[d][0]*x0 + c.zw[d][1]*x1);
    }
    float dt_raw = 0.0f; Bs = 0.0f; Cs = 0.0f;
#pragma unroll
    for (int j = 0; j < DI; ++j) {
        dt_raw += c.xp_dt[j]*xcv[j];
        Bs     += c.xp_B[j]*xcv[j];
        Cs     += c.xp_C[j]*xcv[j];
    }
    dtv0 = softplus_f(c.dtw0*dt_raw + c.dtb0);
    dtv1 = softplus_f(c.dtw1*dt_raw + c.dtb1);
}

// ---------------------------------------------------------------------------
// Phase 1: per-chunk P = prod(dA), q = h(end | h0 = 0).  Exact linear-scan
// decomposition; chunks run fully parallel (grid = NCHUNK x B, 1 wave each).
// ---------------------------------------------------------------------------
__global__ void __launch_bounds__(32) mamba_chunk_pq(
        const float* __restrict__ x,        const float* __restrict__ in_proj_w,
        const float* __restrict__ conv_w,   const float* __restrict__ conv_b,
        const float* __restrict__ x_proj_w, const float* __restrict__ dt_proj_w,
        const float* __restrict__ dt_proj_b,const float* __restrict__ A_log,
        float* __restrict__ Pbuf, float* __restrict__ Qbuf, int L, int CH)
{
    const int chunk = blockIdx.x, b = blockIdx.y;
    const int lane = threadIdx.x;
    LaneCtx c;
    lane_init(c, lane, in_proj_w, conv_w, conv_b, x_proj_w, dt_proj_w, dt_proj_b, A_log);

    const float* xb0 = x + (size_t)b * DM * L;
    const float* xb1 = xb0 + L;
    const int t0 = chunk * CH;
    const int t1 = min(L, t0 + CH);

    float hist[DI][3];
    init_hist(c, xb0, xb1, t0, hist);

    float P0 = 1.0f, P1 = 1.0f, h0 = 0.0f, h1 = 0.0f;
    float xcv[DI], zsil[DI], dtv0, dtv1, Bs, Cs;
    for (int t = t0; t < t1; ++t) {
        step_front<false>(c, xb0[t], xb1[t], hist, xcv, zsil, dtv0, dtv1, Bs, Cs);
        const float dA0 = expf(dtv0 * c.A0);
        const float dA1 = expf(dtv1 * c.A1);
        P0 *= dA0; P1 *= dA1;
        h0 = dA0*h0 + dtv0*Bs*xcv[c.d0];
        h1 = dA1*h1 + dtv1*Bs*xcv[c.d1];
    }
    const size_t base = ((size_t)b * NCHUNK + chunk) * (DI*DS);
    Pbuf[base + c.d0*DS + c.s] = P0;
    Pbuf[base + c.d1*DS + c.s] = P1;
    Qbuf[base + c.d0*DS + c.s] = h0;
    Qbuf[base + c.d1*DS + c.s] = h1;
}

// ---------------------------------------------------------------------------
// Phase 2: sequential combine across chunks (128 steps, parallel over B x 64
// state elements). Rewrites Qbuf[chunk] with the INCOMING h0 of that chunk.
// ---------------------------------------------------------------------------
__global__ void __launch_bounds__(64) mamba_combine(
        const float* __restrict__ Pbuf, float* __restrict__ Qbuf)
{
    const int b = blockIdx.x;
    const int e = threadIdx.x; // element (d*16 + s)
    float h = 0.0f;
    for (int i = 0; i < NCHUNK; ++i) {
        const size_t idx = ((size_t)b * NCHUNK + i) * (DI*DS) + e;
        const float Pv = Pbuf[idx];
        const float Qv = Qbuf[idx];
        Qbuf[idx] = h;          // h0 entering chunk i
        h = Pv*h + Qv;          // h leaving chunk i
    }
}

// ---------------------------------------------------------------------------
// Phase 3: replay each chunk with correct h0, emit feat[b][t] =
//   sum_d (y[d] + xc[d]*Dp[d]) * silu(z[d]) * out_proj_w[0][d]
// y-reduction = 16-lane xor-butterfly + one offset-16 shuffle (wave32 native).
// ---------------------------------------------------------------------------
__global__ void __launch_bounds__(32) mamba_chunk_feat(
        const float* __restrict__ x,        const float* __restrict__ in_proj_w,
        const float* __restrict__ conv_w,   const float* __restrict__ conv_b,
        const float* __restrict__ x_proj_w, const float* __restrict__ dt_proj_w,
        const float* __restrict__ dt_proj_b,const float* __restrict__ A_log,
        const float* __restrict__ Dp,       const float* __restrict__ out_proj_w,
        const float* __restrict__ Qbuf,     float* __restrict__ feat, int L, int CH)
{
    const int chunk = blockIdx.x, b = blockIdx.y;
    const int lane = threadIdx.x;
    LaneCtx c;
    lane_init(c, lane, in_proj_w, conv_w, conv_b, x_proj_w, dt_proj_w, dt_proj_b, A_log);

    const float Dp0 = Dp[c.d0], Dp1 = Dp[c.d1];
    const float ow0 = out_proj_w[c.d0], ow1 = out_proj_w[c.d1]; // row 0

    const float* xb0 = x + (size_t)b * DM * L;
    const float* xb1 = xb0 + L;
    float* fb = feat + (size_t)b * L;
    const int t0 = chunk * CH;
    const int t1 = min(L, t0 + CH);

    float hist[DI][3];
    init_hist(c, xb0, xb1, t0, hist);

    const size_t base = ((size_t)b * NCHUNK + chunk) * (DI*DS);
    float h0 = Qbuf[base + c.d0*DS + c.s];
    float h1 = Qbuf[base + c.d1*DS + c.s];

    float xcv[DI], zsil[DI], dtv0, dtv1, Bs, Cs;
    for (int t = t0; t < t1; ++t) {
        step_front<true>(c, xb0[t], xb1[t], hist, xcv, zsil, dtv0, dtv1, Bs, Cs);
        const float dA0 = expf(dtv0 * c.A0);
        const float dA1 = expf(dtv1 * c.A1);
        h0 = dA0*h0 + dtv0*Bs*xcv[c.d0];
        h1 = dA1*h1 + dtv1*Bs*xcv[c.d1];
        float p0 = h0*Cs, p1 = h1*Cs;
#pragma unroll
        for (int off = 8; off >= 1; off >>= 1) {
            p0 += __shfl_xor(p0, off, 32);
            p1 += __shfl_xor(p1, off, 32);
        }
        float u = (p0 + xcv[c.d0]*Dp0)*zsil[c.d0]*ow0
                + (p1 + xcv[c.d1]*Dp1)*zsil[c.d1]*ow1;
        u += __shfl_xor(u, 16, 32);
        if (lane == 0) fb[t] = u;
    }
}

// ---------------------------------------------------------------------------
// K-split feat(32 x L) @ proj_w.T(L x 512) via V_WMMA_F32_16X16X4_F32.
// Grid = (N/16, M/16, KSPLIT); one wave per (tile, k-slice) -> 1024 waves,
// each accumulating a partial 16x16 fp32 tile over ~L/KSPLIT of K.
// A layout (16x4 f32): lanes 0-15 = rows M0..15, {vgpr0,vgpr1} = {K0,K1};
//                      lanes 16-31 same rows,   {vgpr0,vgpr1} = {K2,K3}.
// B layout mirrors it (rows striped across lanes). C/D: vgpr r -> row r/r+8.
// ---------------------------------------------------------------------------
__global__ void __launch_bounds__(32) proj_wmma_kernel(
        const float* __restrict__ feat, const float* __restrict__ proj_w,
        float* __restrict__ pbuf, int L, int KC)
{
    const int lane = threadIdx.x;
    const int half = lane >> 4;        // 0: K0/K1, 1: K2/K3
    const int l15  = lane & 15;
    const int n0 = blockIdx.x * 16;    // output column tile (proj rows)
    const int m0 = blockIdx.y * 16;    // output row tile (batch)
    const int slice = blockIdx.z;      // K-slice

    const int k_beg = slice * KC;
    const int k_end = min(L, k_beg + KC);

    const float* arow = feat   + (size_t)(m0 + l15) * L; // A[m][*]
    const float* brow = proj_w + (size_t)(n0 + l15) * L; // B[k][n] = proj_w[n][k]

    v8f acc = {};
    for (int k = k_beg; k < k_end; k += 4) {
        const int kk = k + half*2;
        v2f a, bm;
        a.x  = arow[kk];  a.y  = arow[kk + 1];
        bm.x = brow[kk];  bm.y = brow[kk + 1];
        __builtin_prefetch(arow + kk + 1024, 0, 0); // -> global_prefetch_b8
        __builtin_prefetch(brow + kk + 1024, 0, 0);
        acc = __builtin_amdgcn_wmma_f32_16x16x4_f32(
                  /*neg_a=*/false, a, /*neg_b=*/false, bm,
                  /*c_mod=*/(short)0, acc, /*reuse_a=*/false, /*reuse_b=*/false);
    }
    const int col  = n0 + l15;
    const int row0 = m0 + half*8;
    float* pslice = pbuf + (size_t)slice * (2*16) * NPROJ; // 32 x 512 partial
#pragma unroll
    for (int r = 0; r < 8; ++r)
        pslice[(size_t)(row0 + r) * NPROJ + col] = acc[r];
}

// Fixup reduction: hmat[m][n] = sum over KSPLIT partials (exact fp32 adds).
__global__ void __launch_bounds__(256) proj_reduce_kernel(
        const float* __restrict__ pbuf, float* __restrict__ hmat, int MN)
{
    const int i = blockIdx.x * blockDim.x + threadIdx.x;
    if (i < MN) {
        float acc = 0.0f;
#pragma unroll
        for (int s = 0; s < KSPLIT; ++s)
            acc += pbuf[(size_t)s * MN + i];
        hmat[i] = acc;
    }
}

// Classifier head: logit[b] = cls_b + sum_n cls_w[n]*(hmat[b][n]+proj_b[n]); sigmoid.
__global__ void cls_kernel(const float* __restrict__ hmat,
                           const float* __restrict__ proj_b,
                           const float* __restrict__ cls_w,
                           const float* __restrict__ cls_b,
                           float* __restrict__ out, int B)
{
    const int b = threadIdx.x;
    if (b < B) {
        float acc = cls_b[0];
        for (int n = 0; n < NPROJ; ++n)
            acc += cls_w[n] * (hmat[(size_t)b*NPROJ + n] + proj_b[n]);
        out[b] = 1.0f / (1.0f + expf(-acc));
    }
}

extern "C" void kernel_launch(void* const* d_in, const int* in_sizes, int n_in,
                              void* d_out, int out_size, void* d_ws, size_t ws_size,
                              hipStream_t stream)
{
    const float* x        = (const float*)d_in[0];
    const float* in_proj  = (const float*)d_in[1];
    const float* conv_w   = (const float*)d_in[2];
    const float* conv_b   = (const float*)d_in[3];
    const float* x_proj   = (const float*)d_in[4];
    const float* dt_w     = (const float*)d_in[5];
    const float* dt_b     = (const float*)d_in[6];
    const float* A_log    = (const float*)d_in[7];
    const float* Dp       = (const float*)d_in[8];
    const float* out_proj = (const float*)d_in[9];
    const float* proj_w   = (const float*)d_in[10];
    const float* proj_b   = (const float*)d_in[11];
    const float* cls_w    = (const float*)d_in[12];
    const float* cls_b    = (const float*)d_in[13];

    const int L  = in_sizes[10] / NPROJ;     // SEQ_LEN from proj_w (512, L)
    const int B  = in_sizes[0] / (DM * L);   // BATCH from x (B, 2, L)
    const int CH = (L + NCHUNK - 1) / NCHUNK;
    // K-slice length: multiple of 4 (WMMA K-step) covering ceil(L/4/KSPLIT) groups
    const int KC = ((L/4 + KSPLIT - 1) / KSPLIT) * 4;
    const int MN = B * NPROJ;

    // Workspace: feat | Pbuf | Qbuf | hmat | pbuf  (~9 MB total)
    float* feat = (float*)d_ws;
    float* Pbuf = feat + (size_t)B * L;
    float* Qbuf = Pbuf + (size_t)B * NCHUNK * DI * DS;
    float* hmat = Qbuf + (size_t)B * NCHUNK * DI * DS;
    float* pbuf = hmat + (size_t)MN;

    dim3 gchunk(NCHUNK, B);
    mamba_chunk_pq<<<gchunk, 32, 0, stream>>>(x, in_proj, conv_w, conv_b, x_proj,
                                              dt_w, dt_b, A_log, Pbuf, Qbuf, L, CH);
    mamba_combine<<<B, 64, 0, stream>>>(Pbuf, Qbuf);
    mamba_chunk_feat<<<gchunk, 32, 0, stream>>>(x, in_proj, conv_w, conv_b, x_proj,
                                                dt_w, dt_b, A_log, Dp, out_proj,
                                                Qbuf, feat, L, CH);
    proj_wmma_kernel<<<dim3(NPROJ/16, B/16, KSPLIT), 32, 0, stream>>>(feat, proj_w,
                                                                      pbuf, L, KC);
    proj_reduce_kernel<<<(MN + 255)/256, 256, 0, stream>>>(pbuf, hmat, MN);
    cls_kernel<<<1, B, 0, stream>>>(hmat, proj_b, cls_w, cls_b, (float*)d_out, B);
}